// PlanningLoss_21397527069385
// MI455X (gfx1250) — compile-verified
//
#include <hip/hip_runtime.h>
#include <hip/hip_bf16.h>

// ---------------------------------------------------------------------------
// PlanningLoss for MI455X (gfx1250, wave32).
// Memory-bound: 252 MB of BEV masks dominates (~10.8 us at 23.3 TB/s floor).
// Per-batch block: move both 48x80 f32 masks (15 KB each, contiguous) into
// LDS with the Tensor Data Mover (tensor_load_to_lds, TENSORcnt), compute the
// clamped window max/mean from LDS, write 5 partial sums per batch. A single
// wave32 reduce kernel sums the B x 5 partials with V_WMMA_F32_16X16X4_F32
// (ones-matrix column-sum trick) and applies the loss weights.
// ---------------------------------------------------------------------------

#define BEV_H 48
#define BEV_W 80
#define MASK_ELEMS (BEV_H * BEV_W)   // 3840
#define S_STEPS 6
#define MPPX 0.5f
#define MPPY 0.625f
#define CX_I 40
#define CY_I 24
#define COL_HW 4
#define COL_HL 5
#define LANE_HW 2
#define LANE_HL 3

typedef __attribute__((ext_vector_type(2))) float v2f;
typedef __attribute__((ext_vector_type(8))) float v8f;
typedef __attribute__((ext_vector_type(4))) unsigned int u32x4;
typedef __attribute__((ext_vector_type(4))) int i32x4;
typedef __attribute__((ext_vector_type(8))) int i32x8;

// ---------------------------------------------------------------------------
// Wave32 cross-lane sum via the f32 WMMA. A = ones(16x4); each lane holds
// {v, 0} of B (4x16). Every row of D equals the column sums of B, so each
// lane's c[0] = colsum[n]; the 32-lane xor-add double counts -> *0.5.
// EXEC must be all ones at call sites (full wave, no divergence).
// ---------------------------------------------------------------------------
__device__ __forceinline__ float wave32_sum_wmma(float v) {
  v2f a; a[0] = 1.0f; a[1] = 1.0f;
  v2f b; b[0] = v;    b[1] = 0.0f;
  v8f c = {};
  c = __builtin_amdgcn_wmma_f32_16x16x4_f32(
      /*neg_a=*/false, a, /*neg_b=*/false, b,
      /*c_mod=*/(short)0, c, /*reuse_a=*/false, /*reuse_b=*/false);
  float r = c[0];
  r += __shfl_xor(r, 16, 32);
  r += __shfl_xor(r, 8, 32);
  r += __shfl_xor(r, 4, 32);
  r += __shfl_xor(r, 2, 32);
  r += __shfl_xor(r, 1, 32);
  return r * 0.5f;
}

// ---------------------------------------------------------------------------
// TDM: DMA one contiguous run of `MASK_ELEMS` f32 from global to LDS.
// D# layout per CDNA5 ISA ch.8 (group0 128b, group1 256b); 1-row 2D tile:
//   tensor_dim0 = tile_dim0 = MASK_ELEMS, tensor_dim1 = tile_dim1 = 1,
//   data_size = 4 bytes (code 2), stride = MASK_ELEMS.
// Generic LDS pointers carry the LDS byte offset in their low 32 bits.
// ---------------------------------------------------------------------------
#if __has_builtin(__builtin_amdgcn_tensor_load_to_lds)
#define USE_TDM 1
__device__ __forceinline__ void tdm_load_mask(const float* gsrc, float* ldst) {
  const unsigned long long ga = (unsigned long long)(uintptr_t)gsrc;
  const unsigned int la = (unsigned int)(uintptr_t)ldst;

  u32x4 g0;
  g0[0] = 1u;                                      // count=1 (valid user D#)
  g0[1] = la;                                      // lds_addr [63:32]
  g0[2] = (unsigned int)(ga & 0xFFFFFFFFu);        // global_addr[31:0]
  g0[3] = (unsigned int)((ga >> 32) & 0x01FFFFFFu) // global_addr[56:32]
          | (2u << 30);                            // type=2 ("image")

  i32x8 g1;
  g1[0] = (int)(2u << 16);                         // data_size=4B, wg_mask=0
  g1[1] = (int)((MASK_ELEMS & 0xFFFFu) << 16);     // tensor_dim0[15:0] @63:48
  g1[2] = (int)(((MASK_ELEMS >> 16) & 0xFFFFu)     // tensor_dim0[31:16] @79:64
                | (1u << 16));                     // tensor_dim1=1      @95:80
  g1[3] = (int)((MASK_ELEMS & 0xFFFFu) << 16);     // tile_dim0 @127:112
  g1[4] = 1;                                       // tile_dim1=1, tile_dim2=0
  g1[5] = MASK_ELEMS;                              // tensor_dim0_stride[31:0]
  g1[6] = 0;                                       // stride[47:32], dim1_stride lo
  g1[7] = 0;

  i32x4 z; z[0] = 0; z[1] = 0; z[2] = 0; z[3] = 0;
#if __clang_major__ >= 23
  i32x8 z8; z8[0]=0; z8[1]=0; z8[2]=0; z8[3]=0; z8[4]=0; z8[5]=0; z8[6]=0; z8[7]=0;
  __builtin_amdgcn_tensor_load_to_lds(g0, g1, z, z, z8, 0);
#else
  __builtin_amdgcn_tensor_load_to_lds(g0, g1, z, z, 0);
#endif
}
#else
#define USE_TDM 0
#endif

// ---------------------------------------------------------------------------
// One block per batch. partials[b*5 + k] = per-batch sums of:
// 0: sum (pred-targ)^2 (12 coords)   1: sum collision risk (6 pts)
// 2: sum lane compliance (6 pts)     3: sum |accel| (4 second diffs)
// 4: sum (conf - (1-risk))^2 (6 pts)
// ---------------------------------------------------------------------------
__global__ __launch_bounds__(128) void planning_batch_kernel(
    const float* __restrict__ pred, const float* __restrict__ conf,
    const float* __restrict__ targ, const float* __restrict__ det,
    const float* __restrict__ driv, float* __restrict__ partials) {
  __shared__ float s_det[MASK_ELEMS];
  __shared__ float s_driv[MASK_ELEMS];
  __shared__ float red[S_STEPS][5];

  const int b = blockIdx.x;
  const int tid = threadIdx.x;
  const size_t moff = (size_t)b * MASK_ELEMS;

#if USE_TDM
  // Wave 0 issues both DMA descriptors (TDM ignores EXEC; waves 1..3 branch
  // around on execz), waits TENSORcnt==0, then everyone syncs.
  if (tid < 32) {
    tdm_load_mask(det + moff, s_det);
    tdm_load_mask(driv + moff, s_driv);
    __builtin_amdgcn_s_wait_tensorcnt(0);
  }
  __syncthreads();
#else
  // Fallback: coalesced float4 stream through VGPRs.
  __builtin_prefetch(det + moff, 0, 0);
  __builtin_prefetch(driv + moff, 0, 0);
  const float4* det4 = (const float4*)(det + moff);
  const float4* driv4 = (const float4*)(driv + moff);
  float4* sd4 = (float4*)s_det;
  float4* sv4 = (float4*)s_driv;
  for (int i = tid; i < MASK_ELEMS / 4; i += 128) {
    sd4[i] = det4[i];
    sv4[i] = driv4[i];
  }
  __syncthreads();
#endif

  if (tid < S_STEPS) {
    const int s = tid;
    const size_t pbase = ((size_t)b * S_STEPS + s) * 2;
    const float px_m = pred[pbase + 0];
    const float py_m = pred[pbase + 1];
    const float tx = targ[pbase + 0];
    const float ty = targ[pbase + 1];
    const float dx = px_m - tx, dy = py_m - ty;
    const float traj_sq = dx * dx + dy * dy;

    // .astype(int32) truncates toward zero (C cast semantics), then clamp.
    int px = (int)(px_m / MPPX + (float)CX_I);
    int py = (int)(py_m / MPPY + (float)CY_I);
    px = min(max(px, 0), BEV_W - 1);
    py = min(max(py, 0), BEV_H - 1);

    // Collision: clamped-window max (mask >= 0, so init 0 is exact).
    float risk = 0.0f;
    {
      const int y0 = max(py - COL_HL, 0), y1 = min(py + COL_HL, BEV_H - 1);
      const int x0 = max(px - COL_HW, 0), x1 = min(px + COL_HW, BEV_W - 1);
      for (int y = y0; y <= y1; ++y) {
        const float* row = s_det + y * BEV_W;
        for (int x = x0; x <= x1; ++x) risk = fmaxf(risk, row[x]);
      }
    }

    // Lane compliance: clamped-window mean.
    float compliance;
    {
      const int y0 = max(py - LANE_HL, 0), y1 = min(py + LANE_HL + 1, BEV_H);
      const int x0 = max(px - LANE_HW, 0), x1 = min(px + LANE_HW + 1, BEV_W);
      float ssum = 0.0f;
      for (int y = y0; y < y1; ++y) {
        const float* row = s_driv + y * BEV_W;
        for (int x = x0; x < x1; ++x) ssum += row[x];
      }
      compliance = ssum / (float)((y1 - y0) * (x1 - x0));
    }

    const float cf = conf[(size_t)b * S_STEPS + s];
    const float cdiff = cf - (1.0f - risk);

    // Smoothness: |p[s+2] - 2 p[s+1] + p[s]| for s in 0..3.
    float sm = 0.0f;
    if (s < S_STEPS - 2) {
      const float p0x = pred[pbase + 0], p0y = pred[pbase + 1];
      const float p1x = pred[pbase + 2], p1y = pred[pbase + 3];
      const float p2x = pred[pbase + 4], p2y = pred[pbase + 5];
      const float ax = p2x - 2.0f * p1x + p0x;
      const float ay = p2y - 2.0f * p1y + p0y;
      sm = sqrtf(ax * ax + ay * ay);
    }

    red[s][0] = traj_sq;
    red[s][1] = risk;
    red[s][2] = compliance;
    red[s][3] = sm;
    red[s][4] = cdiff * cdiff;
  }
  __syncthreads();

  if (tid == 0) {
    float a0 = 0.f, a1 = 0.f, a2 = 0.f, a3 = 0.f, a4 = 0.f;
    for (int s = 0; s < S_STEPS; ++s) {
      a0 += red[s][0]; a1 += red[s][1]; a2 += red[s][2];
      a3 += red[s][3]; a4 += red[s][4];
    }
    float* p = partials + (size_t)b * 5;
    p[0] = a0; p[1] = a1; p[2] = a2; p[3] = a3; p[4] = a4;
  }
}

// Single wave32 reduces nblocks x 5 partials; WMMA does the cross-lane sums.
__global__ __launch_bounds__(32) void planning_reduce_kernel(
    const float* __restrict__ partials, int nblocks, float* __restrict__ out) {
  const int lane = threadIdx.x;
  float a0 = 0.f, a1 = 0.f, a2 = 0.f, a3 = 0.f, a4 = 0.f;
  for (int i = lane; i < nblocks; i += 32) {
    const float* p = partials + (size_t)i * 5;
    a0 += p[0]; a1 += p[1]; a2 += p[2]; a3 += p[3]; a4 += p[4];
  }
  const float t0 = wave32_sum_wmma(a0);
  const float t1 = wave32_sum_wmma(a1);
  const float t2 = wave32_sum_wmma(a2);
  const float t3 = wave32_sum_wmma(a3);
  const float t4 = wave32_sum_wmma(a4);
  if (lane == 0) {
    const float nBS = (float)nblocks * (float)S_STEPS;
    const float traj_loss = t0 / (nBS * 2.0f);
    const float col_loss = t1 / nBS;
    const float lane_mean = t2 / nBS;
    const float smooth_loss = t3 / ((float)nblocks * (float)(S_STEPS - 2));
    const float conf_loss = t4 / nBS;
    out[0] = 1.0f * traj_loss + 5.0f * col_loss +
             2.0f * (1.0f - lane_mean) + 0.5f * smooth_loss +
             0.1f * conf_loss;
  }
}

extern "C" void kernel_launch(void* const* d_in, const int* in_sizes, int n_in,
                              void* d_out, int out_size, void* d_ws, size_t ws_size,
                              hipStream_t stream) {
  const float* pred = (const float*)d_in[0];  // [B, 6, 2]
  const float* conf = (const float*)d_in[1];  // [B, 6]
  const float* targ = (const float*)d_in[2];  // [B, 6, 2]
  const float* det  = (const float*)d_in[3];  // [B, 48, 80]
  const float* driv = (const float*)d_in[4];  // [B, 48, 80]
  float* out = (float*)d_out;
  float* partials = (float*)d_ws;             // B*5 floats (160 KB for B=8192)

  const int Bn = in_sizes[3] / MASK_ELEMS;

  planning_batch_kernel<<<Bn, 128, 0, stream>>>(pred, conf, targ, det, driv,
                                                partials);
  planning_reduce_kernel<<<1, 32, 0, stream>>>(partials, Bn, out);
}